// NGCF_84756884619305
// MI455X (gfx1250) — compile-verified
//
#include <hip/hip_runtime.h>
#include <hip/hip_bf16.h>

#define N_AUTHORS 50000
#define N_PAPERS  150000
#define N_NODES   (N_AUTHORS + N_PAPERS)
#define N_EDGES   6400000
#define D_DIM     64
#define L_LAYERS  3
#define OUT_D     (D_DIM * (L_LAYERS + 1))   // 256
#define NEG_SLOPE 0.01f
#define EPS_NORM  1e-12f

typedef __attribute__((ext_vector_type(2))) float v2f;
typedef __attribute__((ext_vector_type(8))) float v8f;

// ---------------------------------------------------------------------------
// Kernel 1: out[:, 0:64] = concat(author, paper)   (raw ego, layer-0 block)
// ---------------------------------------------------------------------------
__global__ __launch_bounds__(256) void ngcf_init_kernel(
    const float* __restrict__ author, const float* __restrict__ paper,
    float* __restrict__ out) {
  size_t tid = (size_t)blockIdx.x * blockDim.x + threadIdx.x;
  if (tid >= (size_t)N_NODES * D_DIM) return;
  int n = (int)(tid >> 6);
  int d = (int)(tid & 63);
  float v = (n < N_AUTHORS) ? author[(size_t)n * D_DIM + d]
                            : paper[(size_t)(n - N_AUTHORS) * D_DIM + d];
  out[(size_t)n * OUT_D + d] = v;
}

// ---------------------------------------------------------------------------
// Kernel 2: edge-list SpMM  side[r,:] += val * ego[c,:]
// ego lives in out[:, egoOff : egoOff+64] (stride OUT_D).
// 16 threads per edge, each handling a float4 chunk.  This phase is the
// roofline: ~3.3 GB of gather/scatter traffic per layer at 23.3 TB/s.
// ---------------------------------------------------------------------------
__global__ __launch_bounds__(256) void ngcf_spmm_kernel(
    const int* __restrict__ rows, const int* __restrict__ cols,
    const float* __restrict__ vals, const float* __restrict__ outBase,
    int egoOff, float* __restrict__ side) {
  size_t tid = (size_t)blockIdx.x * blockDim.x + threadIdx.x;
  size_t e = tid >> 4;
  if (e >= (size_t)N_EDGES) return;
  int chunk = (int)(tid & 15) << 2;           // 0,4,...,60
  int r = rows[e];
  int c = cols[e];
  float v = vals[e];
  const float4 g = *reinterpret_cast<const float4*>(
      outBase + (size_t)c * OUT_D + egoOff + chunk);
  float* dst = side + (size_t)r * D_DIM + chunk;
  atomicAdd(dst + 0, v * g.x);
  atomicAdd(dst + 1, v * g.y);
  atomicAdd(dst + 2, v * g.z);
  atomicAdd(dst + 3, v * g.w);
}

// ---------------------------------------------------------------------------
// Kernel 3: fused dense layer with f32 WMMA (16x16x4), persistent waves.
//   sum = leaky(side @ W1^T + b1); bi = leaky((ego*side) @ W2^T + b2)
//   out[:, dstOff:dstOff+64] = sum + bi            (UNNORMALIZED)
//
// Each wave owns ONE 16-wide output-column tile (nt = wave&3): the B
// fragments of W1/W2 for that tile stay resident in 64 VGPRs across a
// grid-stride loop over 16-row tiles.  Per tile, ALL 32 A-fragments are
// staged into registers first (one batched load clause, single wait), then
// two independent 16-deep WMMA chains issue back-to-back with the
// ego*side pk_muls co-executing on the VALU.
// ---------------------------------------------------------------------------
__global__ __launch_bounds__(256) void ngcf_layer_kernel(
    float* __restrict__ outBase, const float* __restrict__ side,
    const float* __restrict__ W1, const float* __restrict__ b1,
    const float* __restrict__ W2, const float* __restrict__ b2,
    int egoOff, int dstOff) {
  const int wave = threadIdx.x >> 5;
  const int lane = threadIdx.x & 31;
  // A/B fragment lane mapping for V_WMMA_F32_16X16X4_F32 (ISA 7.12.2):
  //   lanes 0-15 hold K = kb+0, kb+1 ; lanes 16-31 hold K = kb+2, kb+3
  const int mrow = lane & 15;
  const int koff = (lane >> 4) << 1;          // 0 or 2
  const int nt   = wave & 3;                  // column tile owned by this wave
  const int ncol = nt * 16 + mrow;

  // ---- preload loop-invariant weight (B) fragments into registers --------
  // B[k][n] = W[n][k] (W row-major 64x64); contiguous in k -> float2 loads.
  v2f bw1[16], bw2[16];
  const float* w1p = W1 + (size_t)ncol * 64 + koff;
  const float* w2p = W2 + (size_t)ncol * 64 + koff;
#pragma unroll
  for (int ks = 0; ks < 16; ++ks) {
    bw1[ks] = *reinterpret_cast<const v2f*>(w1p + ks * 4);
    bw2[ks] = *reinterpret_cast<const v2f*>(w2p + ks * 4);
  }
  const float bb1 = b1[ncol];
  const float bb2 = b2[ncol];

  // ---- grid-stride over 16-row tiles (two tile streams per block) --------
  const int nTiles = N_NODES / 16;            // 12500, exact
  const int stride = gridDim.x * 2;
  for (int t = blockIdx.x * 2 + (wave >> 2); t < nTiles; t += stride) {
    const int rowbase = t * 16;
    const float* srow = side    + (size_t)(rowbase + mrow) * D_DIM + koff;
    const float* erow = outBase + (size_t)(rowbase + mrow) * OUT_D + egoOff + koff;

    // Stage ALL A-fragments for this tile first: 32 independent
    // global_load_b64s -> one clause, one wait, then a pure WMMA burst.
    v2f as[16], ae[16];
#pragma unroll
    for (int ks = 0; ks < 16; ++ks) {
      as[ks] = *reinterpret_cast<const v2f*>(srow + ks * 4);   // A (side)
      ae[ks] = *reinterpret_cast<const v2f*>(erow + ks * 4);   // ego
    }

    v8f acc1 = {};
    v8f acc2 = {};
#pragma unroll
    for (int ks = 0; ks < 16; ++ks) {
      v2f ap = as[ks] * ae[ks];                                // A (ego*side)
      acc1 = __builtin_amdgcn_wmma_f32_16x16x4_f32(
          false, as[ks], false, bw1[ks], (short)0, acc1, false, false);
      acc2 = __builtin_amdgcn_wmma_f32_16x16x4_f32(
          false, ap, false, bw2[ks], (short)0, acc2, false, false);
    }

#pragma unroll
    for (int j = 0; j < 8; ++j) {
      // C/D layout: VGPR j -> row j (lanes 0-15) or row j+8 (lanes 16-31)
      int row = (lane < 16) ? j : (j + 8);
      float x1 = acc1[j] + bb1;
      x1 = (x1 >= 0.0f) ? x1 : NEG_SLOPE * x1;
      float x2 = acc2[j] + bb2;
      x2 = (x2 >= 0.0f) ? x2 : NEG_SLOPE * x2;
      outBase[(size_t)(rowbase + row) * OUT_D + dstOff + ncol] = x1 + x2;
    }
  }
}

// ---------------------------------------------------------------------------
// Kernel 4: normalize out[:, 64:256] row-blocks in place (blocks 1..3).
// One 32-lane wave per (node, layer) block of 64 floats.
// ---------------------------------------------------------------------------
__global__ __launch_bounds__(256) void ngcf_norm_kernel(float* __restrict__ outBase) {
  size_t wid = (size_t)blockIdx.x * 8 + (threadIdx.x >> 5);
  const int lane = threadIdx.x & 31;
  if (wid >= (size_t)N_NODES * L_LAYERS) return;
  int n   = (int)(wid / L_LAYERS);
  int blk = (int)(wid % L_LAYERS) + 1;
  float* p = outBase + (size_t)n * OUT_D + blk * D_DIM;
  float a = p[lane];
  float b = p[lane + 32];
  float ss = a * a + b * b;
#pragma unroll
  for (int off = 16; off > 0; off >>= 1) ss += __shfl_xor(ss, off, 32);
  float nr = fmaxf(sqrtf(ss), EPS_NORM);
  float inv = 1.0f / nr;
  p[lane]      = a * inv;
  p[lane + 32] = b * inv;
}

// ---------------------------------------------------------------------------
extern "C" void kernel_launch(void* const* d_in, const int* in_sizes, int n_in,
                              void* d_out, int out_size, void* d_ws, size_t ws_size,
                              hipStream_t stream) {
  const float* author = (const float*)d_in[0];
  const float* paper  = (const float*)d_in[1];
  const int*   rows   = (const int*)d_in[2];
  const int*   cols   = (const int*)d_in[3];
  const float* vals   = (const float*)d_in[4];
  const float* W1     = (const float*)d_in[5];
  const float* b1     = (const float*)d_in[6];
  const float* W2     = (const float*)d_in[7];
  const float* b2     = (const float*)d_in[8];
  float* out  = (float*)d_out;
  float* side = (float*)d_ws;                     // N_NODES x 64 accumulator

  {
    size_t total = (size_t)N_NODES * D_DIM;
    ngcf_init_kernel<<<(unsigned)((total + 255) / 256), 256, 0, stream>>>(
        author, paper, out);
  }

  for (int k = 0; k < L_LAYERS; ++k) {
    hipMemsetAsync(side, 0, (size_t)N_NODES * D_DIM * sizeof(float), stream);
    size_t sthreads = (size_t)N_EDGES * 16;
    ngcf_spmm_kernel<<<(unsigned)((sthreads + 255) / 256), 256, 0, stream>>>(
        rows, cols, vals, out, k * D_DIM, side);
    // 512 blocks x 8 waves: each wave keeps its weight tile in registers and
    // streams ~12 row tiles through the matrix pipe.
    ngcf_layer_kernel<<<512, 256, 0, stream>>>(
        out, side,
        W1 + (size_t)k * D_DIM * D_DIM, b1 + (size_t)k * D_DIM,
        W2 + (size_t)k * D_DIM * D_DIM, b2 + (size_t)k * D_DIM,
        k * D_DIM, (k + 1) * D_DIM);
  }

  {
    size_t waves = (size_t)N_NODES * L_LAYERS;
    ngcf_norm_kernel<<<(unsigned)((waves + 7) / 8), 256, 0, stream>>>(out);
  }
}